// BinderEnergyGuidance_56547539419402
// MI455X (gfx1250) — compile-verified
//
#include <hip/hip_runtime.h>
#include <math.h>

typedef __attribute__((ext_vector_type(2))) float v2f;
typedef __attribute__((ext_vector_type(8))) float v8f;

#define BATCH   16
#define NBIND   1024
#define MTGT    8192
#define KTOP    204            // max(1, int(0.2*1024))
#define NTILES  (NBIND / 16)   // 64
#define MTILES  (MTGT / 16)    // 512
#define CLASH   3.0f
#define ATTRACT_SCALE 10.0f
#define REPEL_SCALE   5.0f

// ---------------------------------------------------------------------------
// Zero the per-batch repel accumulators (graph-replay-safe determinism).
// ---------------------------------------------------------------------------
__global__ void beg_init_kernel(float* __restrict__ repel) {
    if (threadIdx.x < BATCH) repel[threadIdx.x] = 0.0f;
}

// ---------------------------------------------------------------------------
// Phase 1: one wave32 per (batch, 16-row tile of N). For each 16-col tile of
// M, compute the 16x16 inner-product block with V_WMMA_F32_16X16X4_F32
// (K=3 coords + 1 zero pad), then d2 = ||x||^2 + ||y||^2 - 2*xy epilogue:
// running per-row min distance (tracked on d^2) and clash^2 accumulation
// using the single-instruction v_sqrt_f32.
// ---------------------------------------------------------------------------
__global__ void __launch_bounds__(32)
beg_dist_kernel(const float* __restrict__ binder,   // [16,1024,3]
                const float* __restrict__ target,   // [8192,3]
                float* __restrict__ min_out,        // [16,1024]
                float* __restrict__ repel_out)      // [16]
{
    const int wid  = blockIdx.x;
    const int b    = wid / NTILES;
    const int n0   = (wid % NTILES) * 16;
    const int lane = threadIdx.x;    // 0..31
    const int half = lane >> 4;      // 0: K=0,1  1: K=2,pad
    const int l16  = lane & 15;

    // A tile (16x4 fp32). ISA 32-bit A layout: M = lane%16;
    // VGPR0 = K0 (lanes 0-15) / K2 (lanes 16-31); VGPR1 = K1 / K3.
    const float* arow = binder + ((size_t)b * NBIND + (n0 + l16)) * 3;
    const float ax = arow[0], ay = arow[1], az = arow[2];
    v2f amat;
    amat.x = half ? az   : ax;
    amat.y = half ? 0.0f : ay;

    // Row squared norms for the rows this lane's C/D registers cover:
    // M = j + 8*half, j = 0..7.
    float rowx2[8];
#pragma unroll
    for (int j = 0; j < 8; ++j) {
        const float* rr = binder + ((size_t)b * NBIND + (n0 + j + 8 * half)) * 3;
        const float x = rr[0], y = rr[1], z = rr[2];
        rowx2[j] = x * x + y * y + z * z;
    }

    float minv2[8];
#pragma unroll
    for (int j = 0; j < 8; ++j) minv2[j] = 3.0e38f;
    float repel = 0.0f;

    for (int mt = 0; mt < MTILES; ++mt) {
        const int m0 = mt * 16;
        // B tile (4x16 fp32), N = lane%16, same K split as A across halves.
        const float* trow = target + (size_t)(m0 + l16) * 3;
        const float tx = trow[0], ty = trow[1], tz = trow[2];
        const float y2 = tx * tx + ty * ty + tz * tz;
        v2f bmat;
        bmat.x = half ? tz   : tx;
        bmat.y = half ? 0.0f : ty;

        v8f c = {};
        c = __builtin_amdgcn_wmma_f32_16x16x4_f32(
                /*neg_a=*/false, amat, /*neg_b=*/false, bmat,
                /*c_mod=*/(short)0, c, /*reuse_a=*/false, /*reuse_b=*/false);

        // C/D layout: VGPR j holds M = j + 8*half, N = lane%16.
#pragma unroll
        for (int j = 0; j < 8; ++j) {
            const float d2 = fmaxf(rowx2[j] + y2 - 2.0f * c[j], 0.0f);
            minv2[j] = fminf(minv2[j], d2);                 // min on d^2 == min on d
            const float d  = __builtin_amdgcn_sqrtf(d2);    // single v_sqrt_f32
            const float cl = fmaxf(CLASH - d, 0.0f);
            repel += cl * cl;
        }
    }

    // Min over M: each lane covered columns == l16 (mod 16); reduce across
    // the 16 lanes of each half (rows differ between halves). sqrt at the end.
#pragma unroll
    for (int j = 0; j < 8; ++j) {
        float v = minv2[j];
        v = fminf(v, __shfl_xor(v, 1, 32));
        v = fminf(v, __shfl_xor(v, 2, 32));
        v = fminf(v, __shfl_xor(v, 4, 32));
        v = fminf(v, __shfl_xor(v, 8, 32));
        minv2[j] = __builtin_amdgcn_sqrtf(v);
    }
    if (l16 == 0) {           // lane 0 -> rows 0..7, lane 16 -> rows 8..15
#pragma unroll
        for (int j = 0; j < 8; ++j)
            min_out[(size_t)b * NBIND + n0 + j + 8 * half] = minv2[j];
    }

    // Full-wave sum of clash^2, one atomic per wave.
    repel += __shfl_xor(repel, 1, 32);
    repel += __shfl_xor(repel, 2, 32);
    repel += __shfl_xor(repel, 4, 32);
    repel += __shfl_xor(repel, 8, 32);
    repel += __shfl_xor(repel, 16, 32);
    if (lane == 0) atomicAdd(&repel_out[b], repel);
}

// ---------------------------------------------------------------------------
// Phase 2: per batch, stable-rank selection of the 204 smallest min-dists,
// mean them, combine with repel.
// ---------------------------------------------------------------------------
__global__ void __launch_bounds__(256)
beg_select_kernel(const float* __restrict__ min_in,   // [16,1024]
                  const float* __restrict__ repel_in, // [16]
                  float* __restrict__ out)            // [16]
{
    __shared__ float v[NBIND];
    __shared__ float partial[256];
    const int b = blockIdx.x;
    const int t = threadIdx.x;

    for (int i = t; i < NBIND; i += 256) v[i] = min_in[(size_t)b * NBIND + i];
    __syncthreads();

    float sum = 0.0f;
    for (int i = t; i < NBIND; i += 256) {
        const float vi = v[i];
        int rank = 0;
        for (int j = 0; j < NBIND; ++j) {
            const float vj = v[j];
            rank += (vj < vi) || (vj == vi && j < i);
        }
        if (rank < KTOP) sum += vi;   // exactly KTOP elements pass
    }
    partial[t] = sum;
    __syncthreads();
    for (int s = 128; s > 0; s >>= 1) {
        if (t < s) partial[t] += partial[t + s];
        __syncthreads();
    }
    if (t == 0) {
        const float attract = partial[0] / (float)KTOP;
        out[b] = ATTRACT_SCALE * attract + REPEL_SCALE * repel_in[b];
    }
}

// ---------------------------------------------------------------------------
extern "C" void kernel_launch(void* const* d_in, const int* in_sizes, int n_in,
                              void* d_out, int out_size, void* d_ws, size_t ws_size,
                              hipStream_t stream) {
    (void)in_sizes; (void)n_in; (void)out_size; (void)ws_size;
    const float* binder = (const float*)d_in[0];   // [16,1024,3]
    const float* target = (const float*)d_in[1];   // [8192,3]
    float* out = (float*)d_out;                    // [16]

    float* minbuf   = (float*)d_ws;                   // 16*1024 floats
    float* repelbuf = minbuf + (size_t)BATCH * NBIND; // 16 floats

    beg_init_kernel<<<1, 32, 0, stream>>>(repelbuf);
    beg_dist_kernel<<<BATCH * NTILES, 32, 0, stream>>>(binder, target, minbuf, repelbuf);
    beg_select_kernel<<<BATCH, 256, 0, stream>>>(minbuf, repelbuf, out);
}